// KNNCluster_29472065585601
// MI455X (gfx1250) — compile-verified
//
#include <hip/hip_runtime.h>

typedef float v2f __attribute__((ext_vector_type(2)));
typedef float v8f __attribute__((ext_vector_type(8)));

#define L1C 4096
#define L2C 4096
#define NBATCH 8
#define CDIM 64
#define KSEL 16

// One wave32 per (batch, 16-query tile).
// Cross products via V_WMMA_F32_16X16X4_F32 chained over C=64 (16 steps).
// Per-query top-16 via register insertion lists, split across lane halves.
__global__ __launch_bounds__(32) void knn_wmma_kernel(const float* __restrict__ coords1,
                                                      const float* __restrict__ coords2,
                                                      float* __restrict__ out)
{
    __shared__ float distT[16 * 16];       // [query_row][ref_col]
    __shared__ float mergeD[16 * 32];      // per-query: two sorted 16-lists (dist)
    __shared__ int   mergeI[16 * 32];      // per-query: two sorted 16-lists (idx)

    const int lane = threadIdx.x;          // 0..31 (wave32)
    const int m    = lane & 15;            // row-within-tile / ref-col / query id
    const int half = lane >> 4;            // 0 or 1 (K-chunk half for A/B frags)
    const int blk  = blockIdx.x;           // n*256 + qtile
    const int n    = blk >> 8;
    const int q0   = (blk & 255) << 4;

    // ---- Preload A fragments: 16 queries x 64 channels, fp32 16x4 WMMA layout.
    // lane<16: rows M=lane, K = {c0, c0+1}; lane>=16: rows M=lane-16, K = {c0+2, c0+3}
    v2f afrag[16];
    {
        const float* aptr = coords2 + ((size_t)(q0 + m) * NBATCH + n) * CDIM + half * 2;
        #pragma unroll
        for (int kc = 0; kc < 16; ++kc)
            afrag[kc] = *(const v2f*)(aptr + kc * 4);
    }

    // ---- Per-lane sorted top-16 (ascending). Lane m covers ref cols 0..7,
    // lane m+16 covers ref cols 8..15 of every tile, both for query (q0+m).
    float bestD[KSEL];
    int   bestI[KSEL];
    #pragma unroll
    for (int t = 0; t < KSEL; ++t) { bestD[t] = 3.0e38f; bestI[t] = 0; }

    const float* bbase = coords1 + ((size_t)m * NBATCH + n) * CDIM + half * 2;

    for (int r0 = 0; r0 < L1C; r0 += 16) {
        const float* bptr = bbase + (size_t)r0 * (NBATCH * CDIM);

        v8f acc = {};
        float x2p = 0.0f;
        #pragma unroll
        for (int kc = 0; kc < 16; ++kc) {
            v2f b = *(const v2f*)(bptr + kc * 4);      // B frag: 4x16 refs, K-chunk kc
            x2p += b.x * b.x + b.y * b.y;              // partial ||ref||^2 (half the c's)
            acc = __builtin_amdgcn_wmma_f32_16x16x4_f32(
                false, afrag[kc], false, b, (short)0, acc, false, false);
        }
        // Combine the two channel-halves held by lanes l and l^16 (same ref col).
        float x2 = x2p + __shfl_xor(x2p, 16, 32);

        // dist' = ||x||^2 - 2*cross  (query-constant ||q||^2 dropped: order-invariant)
        // D layout: lane holds ref col m; query rows = half*8 + v.
        #pragma unroll
        for (int v = 0; v < 8; ++v)
            distT[(half * 8 + v) * 16 + m] = __builtin_fmaf(acc[v], -2.0f, x2);
        __syncthreads();

        // Selection: this lane's query row is m; its ref columns are half*8..half*8+7.
        #pragma unroll
        for (int j = 0; j < 8; ++j) {
            const int col = half * 8 + j;
            const float d = distT[m * 16 + col];
            if (d < bestD[KSEL - 1]) {                 // strict < keeps lower index on ties
                bestD[KSEL - 1] = d;
                bestI[KSEL - 1] = r0 + col;
                #pragma unroll
                for (int t = KSEL - 1; t > 0; --t) {
                    if (bestD[t] < bestD[t - 1]) {
                        float td = bestD[t]; bestD[t] = bestD[t - 1]; bestD[t - 1] = td;
                        int   ti = bestI[t]; bestI[t] = bestI[t - 1]; bestI[t - 1] = ti;
                    }
                }
            }
        }
        __syncthreads();
    }

    // ---- Merge the two sorted 16-lists per query (lane m has cols 0-7 stream,
    // lane m+16 has cols 8-15 stream) and emit final top-16.
    #pragma unroll
    for (int t = 0; t < KSEL; ++t) {
        mergeD[m * 32 + half * 16 + t] = bestD[t];
        mergeI[m * 32 + half * 16 + t] = bestI[t];
    }
    __syncthreads();

    if (lane < 16) {
        const float* Ad = &mergeD[m * 32];
        const float* Bd = &mergeD[m * 32 + 16];
        const int*   Ai = &mergeI[m * 32];
        const int*   Bi = &mergeI[m * 32 + 16];
        int ia = 0, ib = 0;
        const size_t obase = (size_t)(q0 + m) * NBATCH + n;   // (q * N + n)
        const size_t plane = (size_t)L2C * NBATCH;            // stride per k-rank
        const size_t bhalf = (size_t)KSEL * plane;            // batch-id half offset
        #pragma unroll 1
        for (int t = 0; t < KSEL; ++t) {
            const float da = Ad[ia], db = Bd[ib];
            int sel;
            if (da <= db) { sel = Ai[ia]; ++ia; }
            else          { sel = Bi[ib]; ++ib; }
            out[(size_t)t * plane + obase]         = (float)sel;  // clusters (== local idx)
            out[bhalf + (size_t)t * plane + obase] = (float)n;    // batch id
        }
    }
}

extern "C" void kernel_launch(void* const* d_in, const int* in_sizes, int n_in,
                              void* d_out, int out_size, void* d_ws, size_t ws_size,
                              hipStream_t stream) {
    (void)in_sizes; (void)n_in; (void)d_ws; (void)ws_size; (void)out_size;
    const float* coords1 = (const float*)d_in[0];   // (L1, N, C) fp32
    const float* coords2 = (const float*)d_in[1];   // (L2, N, C) fp32
    float* out = (float*)d_out;                     // [K*L2*N clusters | K*L2*N batch]

    dim3 grid(NBATCH * (L2C / 16));  // 2048 blocks
    dim3 block(32);                  // one wave32 each
    knn_wmma_kernel<<<grid, block, 0, stream>>>(coords1, coords2, out);
}